// EndStopping_48146583388474
// MI455X (gfx1250) — compile-verified
//
#include <hip/hip_runtime.h>
#include <stdint.h>

// ---------------------------------------------------------------------------
// EndStopping for MI455X (gfx1250), wave32.
// Memory-bound (~400MB moved => ~17us at 23.3 TB/s); no matmul structure, so
// the CDNA5-specific path used is GLOBAL_LOAD_ASYNC_TO_LDS (ASYNCcnt) for the
// halo-tile staging in the fused stencil kernel. Output streams use NT store
// hints so the thrice-read 64MB input stays resident in the 192MB L2.
// ---------------------------------------------------------------------------

#define B_    16
#define H_    1024
#define W_    1024
#define NPIX  (H_ * W_)        // 1<<20
#define NTOT  (B_ * NPIX)

#define TOPBINS 16384          // top 16 bits of float pattern (values in [0,1))
#define LOWBINS 65536          // low 16 bits

// order-statistic ranks: 0.02*(N-1)=20971.5, 0.98*(N-1)=1027603.5
#define RK_LO0 20971u
#define RK_HI0 1027603u

// workspace layout (bytes)
#define HIST1_OFF   0
#define HIST1_BYTES (B_ * TOPBINS * 4)              // 1 MiB
#define SEL_OFF     (HIST1_OFF + HIST1_BYTES)
#define SEL_BYTES   4096                            // 16*4*(bin,rem) u32
#define VALS_OFF    (SEL_OFF + SEL_BYTES)
#define VALS_BYTES  4096                            // 16*4 f32
#define PAR_OFF     (VALS_OFF + VALS_BYTES)
#define PAR_BYTES   4096                            // 16*(lo,denom) f32
#define HIST2_OFF   (PAR_OFF + PAR_BYTES)
#define HIST2_BYTES (B_ * 4 * LOWBINS * 4)          // 16 MiB
#define WS_ZERO_BYTES (HIST2_OFF + HIST2_BYTES)

// ---------------- gfx1250 async-to-LDS feature detection -------------------
#ifdef __has_builtin
#  if __has_builtin(__builtin_amdgcn_global_load_async_to_lds_b32)
#    define ASYNC_LDS 1
#  endif
#  if __has_builtin(__builtin_amdgcn_s_wait_asynccnt)
#    define HAVE_WAIT_ASYNC_BUILTIN 1
#  endif
#endif

typedef __attribute__((address_space(1))) int g_int;   // global (AS1)
typedef __attribute__((address_space(3))) int l_int;   // LDS (AS3)

__device__ __forceinline__ void tile_load_f32(float* dst_lds, const float* src_g) {
#ifdef ASYNC_LDS
  __builtin_amdgcn_global_load_async_to_lds_b32(
      (g_int*)(uintptr_t)src_g,
      (l_int*)(uint32_t)(uintptr_t)dst_lds,
      0, 0);
#else
  *dst_lds = *src_g;
#endif
}

__device__ __forceinline__ void tile_load_wait() {
#ifdef ASYNC_LDS
#  ifdef HAVE_WAIT_ASYNC_BUILTIN
  __builtin_amdgcn_s_wait_asynccnt(0);
#  else
  asm volatile("s_wait_asynccnt 0" ::: "memory");
#  endif
#endif
}

__device__ __forceinline__ unsigned topbin_of(unsigned key) {
  unsigned b = key >> 16;
  return b < (TOPBINS - 1u) ? b : (TOPBINS - 1u);
}

// ---------------------------------------------------------------------------
__global__ void k_zero(unsigned* __restrict__ w, int n) {
  int i = blockIdx.x * blockDim.x + threadIdx.x;
  int stride = gridDim.x * blockDim.x;
  for (; i < n; i += stride) __builtin_nontemporal_store(0u, &w[i]);
}

// pass 1: 16384-bin histogram of top 16 float bits, LDS-staged.
// grid = 1024 blocks x 256 threads, 16384 elements/block (one batch per block)
__global__ void k_hist1(const float* __restrict__ edge, unsigned* __restrict__ hist1) {
  __shared__ unsigned sh[TOPBINS];           // 64 KiB
  const int t = threadIdx.x;
  for (int i = t; i < TOPBINS; i += 256) sh[i] = 0u;
  __syncthreads();
  const size_t base = (size_t)blockIdx.x * 16384;
  const int b = (int)(base >> 20);
  const float4* p = (const float4*)(edge + base);
  for (int i = 0; i < 16; ++i) {
    float4 v = p[t + i * 256];
    atomicAdd(&sh[topbin_of(__float_as_uint(v.x))], 1u);
    atomicAdd(&sh[topbin_of(__float_as_uint(v.y))], 1u);
    atomicAdd(&sh[topbin_of(__float_as_uint(v.z))], 1u);
    atomicAdd(&sh[topbin_of(__float_as_uint(v.w))], 1u);
  }
  __syncthreads();
  unsigned* gh = hist1 + (size_t)b * TOPBINS;
  for (int i = t; i < TOPBINS; i += 256) {
    unsigned c = sh[i];
    if (c) atomicAdd(&gh[i], c);
  }
}

// pass 1.5: for each (batch, rank j in 0..3) find top-bin and residual rank.
// grid = 64 blocks x 256 threads
__global__ void k_select1(const unsigned* __restrict__ hist1, unsigned* __restrict__ sel) {
  __shared__ unsigned part[256];
  const int blk = blockIdx.x;
  const int b = blk >> 2, j = blk & 3;
  const unsigned rank = (j & 2) ? (RK_HI0 + (unsigned)(j & 1)) : (RK_LO0 + (unsigned)(j & 1));
  const unsigned* h = hist1 + (size_t)b * TOPBINS;
  const int t = threadIdx.x;
  unsigned s = 0;
  for (int k = 0; k < 64; ++k) s += h[t * 64 + k];
  part[t] = s;
  __syncthreads();
  if (t == 0) {
    unsigned cum = 0; int chunk = 0; unsigned before = 0;
    for (int i = 0; i < 256; ++i) {
      if (cum + part[i] > rank) { chunk = i; before = cum; break; }
      cum += part[i];
    }
    unsigned c = before; int bin = chunk * 64;
    for (int k = 0; k < 64; ++k) {
      unsigned cnt = h[chunk * 64 + k];
      if (c + cnt > rank) { bin = chunk * 64 + k; before = c; break; }
      c += cnt;
    }
    sel[blk * 2 + 0] = (unsigned)bin;
    sel[blk * 2 + 1] = rank - before;
  }
}

// pass 2: low-16-bit histogram for the 4 selected top bins per batch.
// grid = 16384 blocks x 256 threads, 1024 elements/block
__global__ void k_hist2(const float* __restrict__ edge, const unsigned* __restrict__ sel,
                        unsigned* __restrict__ hist2) {
  const int t = threadIdx.x;
  const size_t base = (size_t)blockIdx.x * 1024;
  const int b = (int)(base >> 20);
  const unsigned sb0 = sel[(b * 4 + 0) * 2];
  const unsigned sb1 = sel[(b * 4 + 1) * 2];
  const unsigned sb2 = sel[(b * 4 + 2) * 2];
  const unsigned sb3 = sel[(b * 4 + 3) * 2];
  float4 v = ((const float4*)(edge + base))[t];
  float vv[4] = {v.x, v.y, v.z, v.w};
#pragma unroll
  for (int i = 0; i < 4; ++i) {
    unsigned key = __float_as_uint(vv[i]);
    unsigned top = topbin_of(key);
    unsigned low = key & 0xFFFFu;
    if (top == sb0) atomicAdd(&hist2[(((size_t)b * 4 + 0) << 16) + low], 1u);
    if (top == sb1) atomicAdd(&hist2[(((size_t)b * 4 + 1) << 16) + low], 1u);
    if (top == sb2) atomicAdd(&hist2[(((size_t)b * 4 + 2) << 16) + low], 1u);
    if (top == sb3) atomicAdd(&hist2[(((size_t)b * 4 + 3) << 16) + low], 1u);
  }
}

// pass 2.5: extract exact order-statistic values. grid = 64 blocks x 256 threads
__global__ void k_select2(const unsigned* __restrict__ hist2, const unsigned* __restrict__ sel,
                          float* __restrict__ vals) {
  __shared__ unsigned part[256];
  const int blk = blockIdx.x;                       // b*4 + j
  const unsigned* h = hist2 + ((size_t)blk << 16);
  const unsigned topbin = sel[blk * 2 + 0];
  const unsigned rank = sel[blk * 2 + 1];
  const int t = threadIdx.x;
  unsigned s = 0;
  for (int k = 0; k < 256; ++k) s += h[t * 256 + k];
  part[t] = s;
  __syncthreads();
  if (t == 0) {
    unsigned cum = 0; int chunk = 0; unsigned before = 0;
    for (int i = 0; i < 256; ++i) {
      if (cum + part[i] > rank) { chunk = i; before = cum; break; }
      cum += part[i];
    }
    unsigned c = before; int low = chunk * 256;
    for (int k = 0; k < 256; ++k) {
      unsigned cnt = h[chunk * 256 + k];
      if (c + cnt > rank) { low = chunk * 256 + k; break; }
      c += cnt;
    }
    vals[blk] = __uint_as_float((topbin << 16) | (unsigned)low);
  }
}

__global__ void k_params(const float* __restrict__ vals, float* __restrict__ par) {
  int b = threadIdx.x;
  if (b < B_) {
    float lo = 0.5f * (vals[b * 4 + 0] + vals[b * 4 + 1]);
    float hi = 0.5f * (vals[b * 4 + 2] + vals[b * 4 + 3]);
    par[2 * b + 0] = lo;
    par[2 * b + 1] = (hi - lo) + 1e-6f;
  }
}

// ---------------------------------------------------------------------------
// Fused main kernel: normalize + sobel/atan2 binning + directional max scan.
// 64x64 output tile, halo 6 -> 76x76 LDS region staged via async-to-LDS.
#define TS   64
#define HALO 6
#define RS   (TS + 2 * HALO)        // 76
#define RSQ  (RS * RS)              // 5776
#define LOADI ((RSQ + 255) / 256)   // 23

__global__ void __launch_bounds__(256)
k_main(const float* __restrict__ edge, const float* __restrict__ par,
       float* __restrict__ out_end, float* __restrict__ out_bins,
       float* __restrict__ out_xn) {
  __shared__ float tile[RSQ];       // 23104 B
  const int t = threadIdx.x;
  const int b = blockIdx.z;
  const int by = blockIdx.y * TS, bx = blockIdx.x * TS;
  const float* img = edge + (size_t)b * NPIX;

  // stage halo tile (zeros outside the image: matches zero-padded conv/rolls)
  for (int i = 0; i < LOADI; ++i) {
    int p = t + i * 256;
    if (p < RSQ) {
      int ly = p / RS, lx = p - ly * RS;
      int gy = by + ly - HALO, gx = bx + lx - HALO;
      float* dst = &tile[p];
      if ((unsigned)gy < (unsigned)H_ && (unsigned)gx < (unsigned)W_) {
        tile_load_f32(dst, img + (size_t)gy * W_ + gx);
      } else {
        *dst = 0.0f;
      }
    }
  }
  tile_load_wait();
  __syncthreads();

  // in-place percentile normalization
  const float lo = par[2 * b + 0];
  const float den = par[2 * b + 1];
  for (int i = 0; i < LOADI; ++i) {
    int p = t + i * 256;
    if (p < RSQ) {
      float v = (tile[p] - lo) / den;
      v = v < 0.f ? 0.f : (v > 1.f ? 1.f : v);
      tile[p] = v;
    }
  }
  __syncthreads();

  // dy/dx tables for dirs {(0,1),(-1,1),(-1,0),(-1,-1),(0,-1),(1,-1),(1,0),(1,1)},
  // nibble-packed as value+1 to keep them in SGPRs (no scratch).
  const unsigned DXP = 0x21000122u;
  const unsigned DYP = 0x22210001u;
  const float PI_4 = 0.7853981633974483f;

  for (int i = 0; i < 16; ++i) {
    int p = t + i * 256;            // 0..4095
    int oy = p >> 6, ox = p & 63;
    const float* c = &tile[(oy + HALO) * RS + (ox + HALO)];
    float xn = c[0];

    // sobel (cross-correlation, zero-padded 'SAME')
    float tl = c[-RS - 1], tc = c[-RS], tr = c[-RS + 1];
    float ml = c[-1],                 mr = c[1];
    float bl = c[RS - 1],  bc = c[RS], br = c[RS + 1];
    float gxv = (tr + 2.f * mr + br) - (tl + 2.f * ml + bl);
    float gyv = (bl + 2.f * bc + br) - (tl + 2.f * tc + tr);

    float ang = atan2f(gyv, gxv);
    if (ang < 0.f) ang += 6.2831855f;           // mod 2*pi
    int bin = (int)floorf(ang / PI_4);
    bin = bin < 0 ? 0 : (bin > 7 ? 7 : bin);
    int d = (bin + 2) & 7;                      // rot = N_BINS/4

    int dx = (int)((DXP >> (4 * d)) & 0xFu) - 1;
    int dy = (int)((DYP >> (4 * d)) & 0xFu) - 1;
    int step = dy * RS + dx;

    float f = 0.f, bk = 0.f;
    const float* pf = c;
    const float* pb = c;
#pragma unroll
    for (int r = 1; r <= 6; ++r) {
      pf -= step; pb += step;
      f  = fmaxf(f,  *pf);
      bk = fmaxf(bk, *pb);
    }
    float mn = fminf(f, bk), mx = fmaxf(f, bk);
    float ratio = mn / (mx + 1e-6f);
    float e = xn * (1.f - ratio);
    e = e < 0.f ? 0.f : (e > 1.f ? 1.f : e);

    size_t o = (size_t)b * NPIX + (size_t)(by + oy) * W_ + (bx + ox);
    // NT stores: outputs are write-once/never-read; keep input resident in L2.
    __builtin_nontemporal_store(e,        &out_end[o]);
    __builtin_nontemporal_store((float)d, &out_bins[o]);
    __builtin_nontemporal_store(xn,       &out_xn[o]);
  }
}

// ---------------------------------------------------------------------------
extern "C" void kernel_launch(void* const* d_in, const int* in_sizes, int n_in,
                              void* d_out, int out_size, void* d_ws, size_t ws_size,
                              hipStream_t stream) {
  const float* edge = (const float*)d_in[0];
  float* out = (float*)d_out;
  float* out_end  = out;
  float* out_bins = out + (size_t)NTOT;
  float* out_xn   = out + (size_t)2 * NTOT;

  unsigned char* w = (unsigned char*)d_ws;
  unsigned* hist1 = (unsigned*)(w + HIST1_OFF);
  unsigned* sel   = (unsigned*)(w + SEL_OFF);
  float*    vals  = (float*)(w + VALS_OFF);
  float*    par   = (float*)(w + PAR_OFF);
  unsigned* hist2 = (unsigned*)(w + HIST2_OFF);

  k_zero<<<2048, 256, 0, stream>>>((unsigned*)w, WS_ZERO_BYTES / 4);
  k_hist1<<<1024, 256, 0, stream>>>(edge, hist1);
  k_select1<<<64, 256, 0, stream>>>(hist1, sel);
  k_hist2<<<16384, 256, 0, stream>>>(edge, sel, hist2);
  k_select2<<<64, 256, 0, stream>>>(hist2, sel, vals);
  k_params<<<1, 16, 0, stream>>>(vals, par);
  dim3 grid(W_ / TS, H_ / TS, B_);
  k_main<<<grid, 256, 0, stream>>>(edge, par, out_end, out_bins, out_xn);
}